// LoRe_41188736369458
// MI455X (gfx1250) — compile-verified
//
#include <hip/hip_runtime.h>
#include <math.h>

#define NUM_FEATURES 256
#define NUM_CLASSES  512
#define NUM_BASIS    64
#define ALPHA        0.01f
#define INV_SCALE    (1.0f/100.0f)

typedef __attribute__((ext_vector_type(2))) float v2f;
typedef __attribute__((ext_vector_type(8))) float v8f;

// ---------------- kernel 0: zero global accumulators ----------------
__global__ void k_init(float* __restrict__ g_sums, float* __restrict__ g_cnt) {
  int i = threadIdx.x;
  if (i < NUM_CLASSES) { g_sums[i] = 0.0f; g_cnt[i] = 0.0f; }
}

// ---------------- kernel 1: row-wise softmax of W [512,64] ----------------
// one wave (32 lanes) per row, 2 elements per lane, shuffle reductions
__global__ void k_softmax(const float* __restrict__ W, float* __restrict__ smW) {
  const int row  = blockIdx.x;
  const int lane = threadIdx.x;            // 0..31 (wave32)
  float a = W[row * NUM_BASIS + lane];
  float b = W[row * NUM_BASIS + lane + 32];
  float m = fmaxf(a, b);
  #pragma unroll
  for (int off = 16; off; off >>= 1) m = fmaxf(m, __shfl_xor(m, off, 32));
  float ea = expf(a - m), eb = expf(b - m);
  float s = ea + eb;
  #pragma unroll
  for (int off = 16; off; off >>= 1) s += __shfl_xor(s, off, 32);
  float inv = 1.0f / s;
  smW[row * NUM_BASIS + lane]      = ea * inv;
  smW[row * NUM_BASIS + lane + 32] = eb * inv;
}

// ---------------- kernel 2: D = smW [512x64] @ V^T [64x256] via f32 WMMA ----
// grid = (32 class-tiles, 16 feature-tiles), one wave per 16x16 output tile.
// K is consumed 4 at a time by V_WMMA_F32_16X16X4_F32 (exact f32 math).
__global__ void k_vw_wmma(const float* __restrict__ smW,
                          const float* __restrict__ Vmat,   // [256,64] row-major
                          float* __restrict__ Dmat) {       // [512,256] class-major
  const int mt   = blockIdx.x;          // class tile
  const int nt   = blockIdx.y;          // feature tile
  const int lane = threadIdx.x;         // 0..31, EXEC all ones (uniform flow)
  const int half = lane >> 4;
  const int l16  = lane & 15;

  v8f acc = {};
  // A-matrix 16x4 f32 layout: lane=M, VGPR v / half h -> K = 2h + v
  const float* aRow = smW  + (size_t)(mt * 16 + l16) * NUM_BASIS + 2 * half;
  // B-matrix 4x16 layout: N = l16, VGPR v / half h -> K = 2h + v; B[k][n] = V[n][k]
  const float* bRow = Vmat + (size_t)(nt * 16 + l16) * NUM_BASIS + 2 * half;

  #pragma unroll
  for (int kb = 0; kb < NUM_BASIS; kb += 4) {
    v2f a; a.x = aRow[kb]; a.y = aRow[kb + 1];
    v2f b; b.x = bRow[kb]; b.y = bRow[kb + 1];
    acc = __builtin_amdgcn_wmma_f32_16x16x4_f32(
        /*neg_a=*/false, a, /*neg_b=*/false, b,
        /*c_mod=*/(short)0, acc, /*reuse_a=*/false, /*reuse_b=*/false);
  }

  // D 16x16 f32 layout: VGPR v / half h -> M = v + 8h, N = l16
  #pragma unroll
  for (int v = 0; v < 8; ++v) {
    int c = mt * 16 + v + 8 * half;
    Dmat[(size_t)c * NUM_FEATURES + nt * 16 + l16] = acc[v];
  }
}

// ---------------- kernel 3: HBM-bound streaming pass ----------------
// One wave per row: 256-f32 dot(X[n], D[class[n]]), log_sigmoid, LDS-atomic
// segment accumulation (ids are sorted -> each block touches few classes),
// sparse flush to global atomics.
__global__ void __launch_bounds__(256)
k_main(const float* __restrict__ X, const int* __restrict__ seg,
       const float* __restrict__ Dmat,
       float* __restrict__ g_sums, float* __restrict__ g_cnt,
       int nrows, int rpb) {
  __shared__ float ls[NUM_CLASSES];
  __shared__ float lc[NUM_CLASSES];
  for (int i = threadIdx.x; i < NUM_CLASSES; i += blockDim.x) { ls[i] = 0.0f; lc[i] = 0.0f; }
  __syncthreads();

  const int wave = threadIdx.x >> 5;     // 8 waves per block
  const int lane = threadIdx.x & 31;
  const int row0   = blockIdx.x * rpb;
  const int rowEnd = (row0 + rpb < nrows) ? (row0 + rpb) : nrows;

  for (int n = row0 + wave; n < rowEnd; n += 8) {
    if (n + 8 < rowEnd)  // prefetch next row's slice -> global_prefetch_b8
      __builtin_prefetch(X + (size_t)(n + 8) * NUM_FEATURES + lane * 8, 0, 3);

    const int c = seg[n];
    const float4* xp = (const float4*)(X    + (size_t)n * NUM_FEATURES) + lane * 2;
    const float4* dp = (const float4*)(Dmat + (size_t)c * NUM_FEATURES) + lane * 2;
    float4 x0 = xp[0], x1 = xp[1];
    float4 d0 = dp[0], d1 = dp[1];
    float p = fmaf(x0.x, d0.x, fmaf(x0.y, d0.y, fmaf(x0.z, d0.z, x0.w * d0.w)))
            + fmaf(x1.x, d1.x, fmaf(x1.y, d1.y, fmaf(x1.z, d1.z, x1.w * d1.w)));
    #pragma unroll
    for (int off = 16; off; off >>= 1) p += __shfl_xor(p, off, 32);

    if (lane == 0) {
      float logit = p * INV_SCALE;
      // stable log_sigmoid: min(x,0) - log1p(exp(-|x|))
      float ll = fminf(logit, 0.0f) - log1pf(expf(-fabsf(logit)));
      atomicAdd(&ls[c], ll);      // ds_add_f32
      atomicAdd(&lc[c], 1.0f);
    }
  }
  __syncthreads();

  for (int i = threadIdx.x; i < NUM_CLASSES; i += blockDim.x) {
    float cv = lc[i];
    if (cv != 0.0f) {
      atomicAdd(&g_sums[i], ls[i]);
      atomicAdd(&g_cnt[i], cv);
    }
  }
}

// ---------------- kernel 4: finalize (nll, reg) ----------------
__global__ void k_finalize(const float* __restrict__ g_sums, const float* __restrict__ g_cnt,
                           const float* __restrict__ Vmat, const float* __restrict__ V_sft,
                           float* __restrict__ out) {
  __shared__ float red[256];
  const int t = threadIdx.x;

  float nl = 0.0f;
  for (int c = t; c < NUM_CLASSES; c += 256)
    nl += g_sums[c] / fmaxf(g_cnt[c], 1.0f);
  red[t] = nl; __syncthreads();
  for (int s = 128; s; s >>= 1) { if (t < s) red[t] += red[t + s]; __syncthreads(); }
  float nll = -red[0];
  __syncthreads();

  float r = 0.0f;
  for (int i = t; i < NUM_FEATURES * NUM_BASIS; i += 256) {
    float d = Vmat[i] - V_sft[i / NUM_BASIS];
    r = fmaf(d, d, r);
  }
  red[t] = r; __syncthreads();
  for (int s = 128; s; s >>= 1) { if (t < s) red[t] += red[t + s]; __syncthreads(); }

  if (t == 0) { out[0] = nll; out[1] = ALPHA * red[0]; }
}

extern "C" void kernel_launch(void* const* d_in, const int* in_sizes, int n_in,
                              void* d_out, int out_size, void* d_ws, size_t ws_size,
                              hipStream_t stream) {
  const float* X     = (const float*)d_in[0];   // [N,256]
  const int*   seg   = (const int*)  d_in[1];   // [N]
  const float* V_sft = (const float*)d_in[2];   // [256]
  const float* W     = (const float*)d_in[3];   // [512,64]
  const float* Vmat  = (const float*)d_in[4];   // [256,64]
  const int nrows = in_sizes[1];

  float* ws     = (float*)d_ws;
  float* smW    = ws;                                   // 512*64
  float* Dmat   = smW  + NUM_CLASSES * NUM_BASIS;       // 512*256
  float* g_sums = Dmat + NUM_CLASSES * NUM_FEATURES;    // 512
  float* g_cnt  = g_sums + NUM_CLASSES;                 // 512
  float* out    = (float*)d_out;

  k_init<<<1, NUM_CLASSES, 0, stream>>>(g_sums, g_cnt);
  k_softmax<<<NUM_CLASSES, 32, 0, stream>>>(W, smW);
  k_vw_wmma<<<dim3(NUM_CLASSES / 16, NUM_FEATURES / 16), 32, 0, stream>>>(smW, Vmat, Dmat);

  const int nblocks = 1024;
  const int rpb = (nrows + nblocks - 1) / nblocks;
  k_main<<<nblocks, 256, 0, stream>>>(X, seg, Dmat, g_sums, g_cnt, nrows, rpb);

  k_finalize<<<1, 256, 0, stream>>>(g_sums, g_cnt, Vmat, V_sft, out);
}